// FCOSPostProcessor_22986664968463
// MI455X (gfx1250) — compile-verified
//
#include <hip/hip_runtime.h>
#include <hip/hip_bf16.h>
#include <stdint.h>

// ---------------- problem constants ----------------
#define N_IMG   8
#define C_CLS   10
#define HW_DIM  262144              // 512*512
#define TOTAL   (C_CLS * HW_DIM)    // 2,621,440 scores per image
#define K_TOP   1000
#define NBINS   4096
#define THRESH_RAW (-2.9444389791664403f)   // logit(0.05): sigmoid(s)>0.05 <=> s>this
#define PI_F    3.14159265358979323846f

// ---------------- streaming config ----------------
#define BLK             256
#define ITERS           16
#define ELEMS_PER_BLOCK (ITERS * BLK * 4)     // 16384
#define BLOCKS_PER_IMG  (TOTAL / ELEMS_PER_BLOCK)  // 160

// histogram replication (cut intra-wave same-address ds_add serialization)
#define HCOPIES  4
#define HSTRIDE  (NBINS + 16)   // +16 words so replicas of a bin sit on different banks

// ---------------- workspace layout (uint32 units) ----------------
#define CAP_AB   1024
#define CAP_TIE  3072
#define OFF_HIST       0
#define OFF_CNT_ABOVE  (OFF_HIST + N_IMG * NBINS)     // 32768
#define OFF_CNT_TIE    (OFF_CNT_ABOVE + N_IMG)        // 32776
#define OFF_THRESH     (OFF_CNT_TIE + N_IMG)          // 32784
#define OFF_ZERO_END   (OFF_THRESH + N_IMG)           // 32792
#define OFF_AB_KEY     (OFF_ZERO_END)
#define OFF_AB_IDX     (OFF_AB_KEY + N_IMG * CAP_AB)
#define OFF_TIE_KEY    (OFF_AB_IDX + N_IMG * CAP_AB)
#define OFF_TIE_IDX    (OFF_TIE_KEY + N_IMG * CAP_TIE)
#define OFF_SEL        (OFF_TIE_IDX + N_IMG * CAP_TIE)
// total ~106k words (~425 KB) of d_ws

// ---------------- output layout (float32 units, concatenated in return order) --
#define OUT_BBOX   0                      // (8,1000,7)
#define OUT_CLS    (N_IMG * K_TOP * 7)    // 56000
#define OUT_LABEL  (OUT_CLS + N_IMG * K_TOP)
#define OUT_SC2    (OUT_LABEL + N_IMG * K_TOP)
#define OUT_VALID  (OUT_SC2 + N_IMG * K_TOP * 2)

// ---------------- helpers ----------------
__device__ __forceinline__ uint32_t fkey(float f) {
    uint32_t u = __float_as_uint(f);
    return u ^ ((uint32_t)((int32_t)u >> 31) | 0x80000000u);  // order-preserving
}
__device__ __forceinline__ float funkey(uint32_t k) {
    uint32_t u = (k & 0x80000000u) ? (k ^ 0x80000000u) : ~k;
    return __uint_as_float(u);
}
__device__ __forceinline__ float sigmoidf(float x) { return 1.0f / (1.0f + expf(-x)); }
__device__ __forceinline__ uint32_t umin_u(uint32_t a, uint32_t b) { return a < b ? a : b; }

// CDNA5 async global->LDS copy (ASYNCcnt-tracked), inline asm per CDNA5 ISA.
// vdst = per-lane LDS byte address, vaddr = per-lane 64-bit global address.
__device__ __forceinline__ void async_cp16(unsigned lds_addr, const void* gaddr) {
    asm volatile("global_load_async_to_lds_b128 %0, %1, off"
                 :: "v"(lds_addr), "v"((unsigned long long)gaddr)
                 : "memory");
}
__device__ __forceinline__ void wait_async_le1() {
    asm volatile("s_wait_asynccnt 0x1" ::: "memory");
}
__device__ __forceinline__ void wait_async_le0() {
    asm volatile("s_wait_asynccnt 0x0" ::: "memory");
}

// ---------------- kernel 0: zero histogram/counters ----------------
__global__ void fcos_zero_ws(uint32_t* __restrict__ ws) {
    int i = blockIdx.x * blockDim.x + threadIdx.x;
    if (i < OFF_ZERO_END) ws[i] = 0u;
}

// ---------------- kernel 1: histogram via async-LDS streaming ----------------
__global__ __launch_bounds__(BLK) void fcos_hist(const float* __restrict__ cls,
                                                 uint32_t* __restrict__ ws) {
    __shared__ uint32_t lhist[HCOPIES * HSTRIDE];     // ~64.3 KB, 4 replicas
    __shared__ __align__(16) float stage[2][BLK * 4]; // 8 KB, double-buffered
    const int tid = threadIdx.x;
    const int n   = blockIdx.x / BLOCKS_PER_IMG;
    const int blk = blockIdx.x % BLOCKS_PER_IMG;

    for (int b = tid; b < HCOPIES * HSTRIDE; b += BLK) lhist[b] = 0u;
    __syncthreads();

    const float4* g = (const float4*)(cls + (size_t)n * TOTAL + (size_t)blk * ELEMS_PER_BLOCK);
    // low 32 bits of a generic LDS pointer == wave-relative LDS byte offset
    unsigned lds0 = (unsigned)(unsigned long long)(&stage[0][tid * 4]);
    unsigned lds1 = (unsigned)(unsigned long long)(&stage[1][tid * 4]);
    uint32_t* myh = &lhist[(tid & (HCOPIES - 1)) * HSTRIDE];

    async_cp16(lds0, (const void*)(g + tid));
    for (int i = 0; i < ITERS; ++i) {
        if (i + 1 < ITERS) {
            async_cp16(((i + 1) & 1) ? lds1 : lds0, (const void*)(g + (i + 1) * BLK + tid));
            wait_async_le1();   // oldest of two outstanding async loads done
        } else {
            wait_async_le0();
        }
        float4 v = *(const float4*)&stage[i & 1][tid * 4];   // ds_load_b128 of own slot
        if (v.x > THRESH_RAW) atomicAdd(&myh[fkey(v.x) >> 20], 1u);
        if (v.y > THRESH_RAW) atomicAdd(&myh[fkey(v.y) >> 20], 1u);
        if (v.z > THRESH_RAW) atomicAdd(&myh[fkey(v.z) >> 20], 1u);
        if (v.w > THRESH_RAW) atomicAdd(&myh[fkey(v.w) >> 20], 1u);
    }
    __syncthreads();
    uint32_t* gh = ws + OFF_HIST + n * NBINS;
    for (int b = tid; b < NBINS; b += BLK) {
        uint32_t c = lhist[b] + lhist[b + HSTRIDE] + lhist[b + 2 * HSTRIDE]
                   + lhist[b + 3 * HSTRIDE];
        if (c) atomicAdd(&gh[b], c);
    }
}

// ---------------- kernel 2: find threshold bin per image (parallel scan) ------
__global__ __launch_bounds__(256) void fcos_thresh(uint32_t* __restrict__ ws) {
    __shared__ uint32_t sc[256];
    __shared__ uint32_t resBin;
    const int n = blockIdx.x, tid = threadIdx.x;
    if (tid == 0) resBin = 0u;   // default: fewer than K candidates overall
    const uint32_t* h = ws + OFF_HIST + n * NBINS;

    // thread t covers ranks r = 16t..16t+15 in DESCENDING bin order (bin = 4095-r)
    uint32_t cbin[16];
    uint32_t csum = 0;
#pragma unroll
    for (int i = 0; i < 16; ++i) {
        uint32_t c = h[4095 - (tid * 16 + i)];
        cbin[i] = c;
        csum += c;
    }
    sc[tid] = csum;
    __syncthreads();
    // Hillis-Steele inclusive scan over the 256 chunk sums
    uint32_t incl = csum;
    for (int off = 1; off < 256; off <<= 1) {
        uint32_t w = (tid >= off) ? sc[tid - off] : 0u;
        __syncthreads();
        incl += w;
        sc[tid] = incl;
        __syncthreads();
    }
    uint32_t excl = incl - csum;  // count of elements in bins strictly above my chunk
    if (excl < K_TOP && excl + csum >= K_TOP) {   // unique crossing chunk
        uint32_t total = excl;
#pragma unroll
        for (int i = 0; i < 16; ++i) {
            uint32_t c = cbin[i];
            if (total + c >= K_TOP) { resBin = 4095u - (uint32_t)(tid * 16 + i); break; }
            total += c;
        }
    }
    __syncthreads();
    if (tid == 0) ws[OFF_THRESH + n] = resBin;
}

// ---------------- kernel 3: compact candidates (L2-resident second pass) ------
__global__ __launch_bounds__(BLK) void fcos_compact(const float* __restrict__ cls,
                                                    uint32_t* __restrict__ ws) {
    const int tid = threadIdx.x;
    const int n   = blockIdx.x / BLOCKS_PER_IMG;
    const int blk = blockIdx.x % BLOCKS_PER_IMG;
    const int base = blk * ELEMS_PER_BLOCK;
    const uint32_t tb = ws[OFF_THRESH + n];
    const float4* g = (const float4*)(cls + (size_t)n * TOTAL + base);
    uint32_t* cntA = ws + OFF_CNT_ABOVE + n;
    uint32_t* cntT = ws + OFF_CNT_TIE + n;
    uint32_t* abK = ws + OFF_AB_KEY + n * CAP_AB;
    uint32_t* abI = ws + OFF_AB_IDX + n * CAP_AB;
    uint32_t* tiK = ws + OFF_TIE_KEY + n * CAP_TIE;
    uint32_t* tiI = ws + OFF_TIE_IDX + n * CAP_TIE;

    for (int i = 0; i < ITERS; ++i) {
        float4 v = g[i * BLK + tid];
        int m0 = base + (i * BLK + tid) * 4;
        float arr[4] = {v.x, v.y, v.z, v.w};
#pragma unroll
        for (int e = 0; e < 4; ++e) {
            float s = arr[e];
            if (s > THRESH_RAW) {
                uint32_t key = fkey(s);
                uint32_t bin = key >> 20;
                if (bin >= tb) {
                    uint32_t m  = (uint32_t)(m0 + e);
                    uint32_t c  = m >> 18;            // channel (HW = 2^18)
                    uint32_t hw = m & (HW_DIM - 1);
                    uint32_t ridx = hw * C_CLS + c;   // reference flat index
                    if (bin > tb) {
                        uint32_t p = atomicAdd(cntA, 1u);
                        if (p < CAP_AB) { abK[p] = key; abI[p] = ridx; }
                    } else {
                        uint32_t p = atomicAdd(cntT, 1u);
                        if (p < CAP_TIE) { tiK[p] = key; tiI[p] = ridx; }
                    }
                }
            }
        }
    }
}

// ---------------- kernel 4: bitonic sort + select top-1000 per image ----------
__global__ __launch_bounds__(1024) void fcos_sort_select(uint32_t* __restrict__ ws,
                                                         float* __restrict__ out) {
    __shared__ unsigned long long buf[4096];   // 32 KB
    const int n = blockIdx.x, tid = threadIdx.x;
    const uint32_t A = umin_u(ws[OFF_CNT_ABOVE + n], CAP_AB);
    const uint32_t T = umin_u(ws[OFF_CNT_TIE + n], CAP_TIE);
    const uint32_t* abK = ws + OFF_AB_KEY + n * CAP_AB;
    const uint32_t* abI = ws + OFF_AB_IDX + n * CAP_AB;
    const uint32_t* tiK = ws + OFF_TIE_KEY + n * CAP_TIE;
    const uint32_t* tiI = ws + OFF_TIE_IDX + n * CAP_TIE;

    for (int i = tid; i < 4096; i += 1024) {
        unsigned long long v = 0ull;   // sentinel: worse than any real candidate
        if (i < (int)A) {
            v = ((unsigned long long)abK[i] << 32) | (uint32_t)(abI[i] ^ 0xFFFFFFFFu);
        } else if (i < (int)(A + T)) {
            int t = i - (int)A;
            v = ((unsigned long long)tiK[t] << 32) | (uint32_t)(tiI[t] ^ 0xFFFFFFFFu);
        }
        buf[i] = v;
    }
    __syncthreads();
    // bitonic sort, descending (key desc, index asc via ~idx in low bits)
    for (int k = 2; k <= 4096; k <<= 1) {
        for (int j = k >> 1; j > 0; j >>= 1) {
            for (int i = tid; i < 4096; i += 1024) {
                int p = i ^ j;
                if (p > i) {
                    unsigned long long a = buf[i], b = buf[p];
                    bool asc = ((i & k) == 0);
                    if (asc ? (a < b) : (a > b)) { buf[i] = b; buf[p] = a; }
                }
            }
            __syncthreads();
        }
    }
    for (int q = tid; q < K_TOP; q += 1024) {
        unsigned long long v = buf[q];
        uint32_t key = (uint32_t)(v >> 32);
        float score, valid;
        uint32_t idx;
        if (key) {
            idx = (uint32_t)v ^ 0xFFFFFFFFu;
            score = sigmoidf(funkey(key));
            valid = 1.0f;
        } else {
            idx = 0u; score = -1.0f; valid = 0.0f;
        }
        int o = n * K_TOP + q;
        out[OUT_CLS + o]   = score;
        out[OUT_LABEL + o] = (float)(idx % C_CLS + 1u);
        out[OUT_VALID + o] = valid;
        ws[OFF_SEL + o] = idx;
    }
}

// ---------------- kernel 5: gather + bbox10d_to_7d ----------------
__device__ __forceinline__ float two_points_angle(float p1x, float p1y, float p2x, float p2y) {
    float at = atanf((p1y - p2y) / (p1x - p2x));
    float q1 = (p1x >= p2x && p1y > p2y) ? 1.f : 0.f;
    float q2 = (p1x <= p2x && p1y > p2y) ? 1.f : 0.f;
    float q3 = (p1x <= p2x && p1y < p2y) ? 1.f : 0.f;
    float q4 = (p1x >= p2x && p1y < p2y) ? 1.f : 0.f;
    return q1 * at + q2 * (PI_F + at) + q3 * (at - PI_F) + q4 * at;
}
__device__ __forceinline__ float angle_limit(float a) {
    return a > PI_F ? a - 2.0f * PI_F : (a < -PI_F ? a + 2.0f * PI_F : a);
}

__global__ void fcos_bbox(const uint32_t* __restrict__ ws,
                          const float* __restrict__ loc,
                          const float* __restrict__ reg,
                          const float* __restrict__ bscore,
                          const float* __restrict__ imgsz,
                          const float* __restrict__ voxel,
                          float* __restrict__ out) {
    int gid = blockIdx.x * blockDim.x + threadIdx.x;
    if (gid >= N_IMG * K_TOP) return;
    int n = gid / K_TOP;
    uint32_t idx = ws[OFF_SEL + gid];
    uint32_t hw = idx / C_CLS;

    float vox  = voxel[0];
    float yoff = imgsz[1] * vox * 0.5f;
    float lx = loc[hw * 2 + 0], ly = loc[hw * 2 + 1];
    const float* r = reg + (size_t)n * C_CLS * HW_DIM + hw;
    float r0 = r[0 * (size_t)HW_DIM], r1 = r[1 * (size_t)HW_DIM];
    float r2 = r[2 * (size_t)HW_DIM], r3 = r[3 * (size_t)HW_DIM];
    float r4 = r[4 * (size_t)HW_DIM], r5 = r[5 * (size_t)HW_DIM];
    float r6 = r[6 * (size_t)HW_DIM], r7 = r[7 * (size_t)HW_DIM];
    float r8 = r[8 * (size_t)HW_DIM], r9 = r[9 * (size_t)HW_DIM];

    float lxv = lx * vox, lyv = ly * vox;
    float x = lxv - (r0 + r2 + r4 + r6) * 0.25f * vox;
    float y = lyv - (r1 + r3 + r5 + r7) * 0.25f * vox - yoff;
    float z = r8 * 0.1f;
    float p1x = lxv - r0 * vox, p1y = lyv - yoff - r1 * vox;
    float p2x = lxv - r2 * vox, p2y = lyv - yoff - r3 * vox;
    float p3x = lxv - r4 * vox, p3y = lyv - yoff - r5 * vox;
    float p4x = lxv - r6 * vox, p4y = lyv - yoff - r7 * vox;
    float w = sqrtf((p3x - p2x) * (p3x - p2x) + (p3y - p2y) * (p3y - p2y));
    float l = sqrtf((p4x - p1x) * (p4x - p1x) + (p4y - p1y) * (p4y - p1y));
    float h = r9 * 0.1f;
    float a1 = two_points_angle(p1x, p1y, p4x, p4y);
    float a2 = angle_limit(two_points_angle(p2x, p2y, p3x, p3y) + PI_F * 0.5f);
    float angle = 0.5f * (a1 + a2);

    float* bb = out + OUT_BBOX + (size_t)gid * 7;
    bb[0] = x; bb[1] = y; bb[2] = z; bb[3] = w; bb[4] = l; bb[5] = h; bb[6] = angle;

    const float* bs = bscore + (size_t)n * 2 * HW_DIM + hw;
    out[OUT_SC2 + gid * 2 + 0] = sigmoidf(bs[0]);
    out[OUT_SC2 + gid * 2 + 1] = sigmoidf(bs[(size_t)HW_DIM]);
}

// ---------------- launch ----------------
extern "C" void kernel_launch(void* const* d_in, const int* in_sizes, int n_in,
                              void* d_out, int out_size, void* d_ws, size_t ws_size,
                              hipStream_t stream) {
    const float* locations  = (const float*)d_in[0];
    const float* box_cls    = (const float*)d_in[1];
    const float* box_reg    = (const float*)d_in[2];
    const float* box_score  = (const float*)d_in[3];
    const float* image_sizes = (const float*)d_in[4];
    const float* voxel_size  = (const float*)d_in[5];
    uint32_t* ws = (uint32_t*)d_ws;
    float* out = (float*)d_out;

    fcos_zero_ws<<<(OFF_ZERO_END + 255) / 256, 256, 0, stream>>>(ws);
    fcos_hist<<<N_IMG * BLOCKS_PER_IMG, BLK, 0, stream>>>(box_cls, ws);
    fcos_thresh<<<N_IMG, 256, 0, stream>>>(ws);
    fcos_compact<<<N_IMG * BLOCKS_PER_IMG, BLK, 0, stream>>>(box_cls, ws);
    fcos_sort_select<<<N_IMG, 1024, 0, stream>>>(ws, out);
    fcos_bbox<<<(N_IMG * K_TOP + 255) / 256, 256, 0, stream>>>(
        ws, locations, box_reg, box_score, image_sizes, voxel_size, out);
}